// Nonlinearity_85770496901716
// MI455X (gfx1250) — compile-verified
//
#include <hip/hip_runtime.h>

#define L_MAX 5
#define NPATHS 69
#define BATCH 32
#define CHANS 64

typedef float v2f __attribute__((ext_vector_type(2)));
typedef float v8f __attribute__((ext_vector_type(8)));

struct KParams {
    const float* x[6];                    // x0..x5
    const float* cg[NPATHS];              // per-path CG tensor
    unsigned long long pbase[NPATHS];     // output base offset (floats) of this path's slab
    unsigned int bstride[NPATHS];         // batch stride (floats) inside this path's out[L]
    int meta[NPATHS];                     // L | l1<<4 | l2<<8
};

// LDS: Y is Kp x ncols (<= 12 x 704), A is 64 x Kp (<= 64 x 12)
#define Y_ELEMS (12 * 704)
#define A_ELEMS (64 * 12)

__global__ __launch_bounds__(256) void cg_tensorprod_wmma(KParams p, float* __restrict__ out) {
    const int path = blockIdx.x;
    const int b    = blockIdx.y;

    const int meta = p.meta[path];
    const int L  = meta & 15;
    const int l1 = (meta >> 4) & 15;
    const int l2 = (meta >> 8) & 15;

    const int K     = 2 * l1 + 1;          // contraction dim (m), 1..11
    const int nl2   = 2 * l2 + 1;          // inner dim (n)
    const int nM    = 2 * L + 1;           // output M dim
    const int ncols = CHANS * nM;          // 64*(2L+1), multiple of 16
    const int nK4   = (K + 3) >> 2;        // WMMA K-chunks of 4
    const int Kp    = nK4 << 2;            // padded K

    const float* __restrict__ x1 = p.x[l1];
    const float* __restrict__ x2 = p.x[l2];
    const float* __restrict__ cg = p.cg[path];

    __shared__ float Ysh[Y_ELEMS];
    __shared__ float Ash[A_ELEMS];

    const int tid = threadIdx.x;

    // ---- Stage A into LDS, zero-padded along K: A[i][k] = x1[b,i,k] ----
    for (int idx = tid; idx < CHANS * Kp; idx += 256) {
        const int i = idx / Kp;
        const int k = idx - i * Kp;
        Ash[idx] = (k < K) ? x1[(size_t)(b * CHANS + i) * K + k] : 0.0f;
    }

    // ---- Build Y in LDS, zero-padded: Y[m][(j,M)] = sum_n cg[M,n,m]*x2[b,j,n] ----
    for (int idx = tid; idx < Kp * ncols; idx += 256) {
        const int m  = idx / ncols;
        const int jM = idx - m * ncols;
        float acc = 0.0f;
        if (m < K) {
            const int j = jM / nM;
            const int M = jM - j * nM;
            const float* __restrict__ xr = x2 + (size_t)(b * CHANS + j) * nl2;
            const float* __restrict__ cr = cg + (size_t)M * nl2 * K + m;
            for (int n = 0; n < nl2; ++n)
                acc += cr[(size_t)n * K] * xr[n];
        }
        Ysh[idx] = acc;
    }
    __syncthreads();

    // ---- WMMA phase: OUT(64 x ncols) = A(64 x Kp) * Y(Kp x ncols) ----
    const int wave = tid >> 5;
    const int lane = tid & 31;
    const int half = lane >> 4;     // lane half: picks K pairs / row groups
    const int n16  = lane & 15;     // column within 16-wide tile / row within A

    const int nColT  = ncols >> 4;          // 16-col tiles
    const int nTiles = 4 * nColT;           // 4 row tiles (64 rows)
    const size_t obase = (size_t)p.pbase[path] + (size_t)b * (size_t)p.bstride[path];

    for (int t = wave; t < nTiles; t += 8) {
        const int ti = t & 3;               // row-tile  (i in [16ti, 16ti+16))
        const int tj = t >> 2;              // col-tile  (jM in [16tj, 16tj+16))

        v8f c = {0.f, 0.f, 0.f, 0.f, 0.f, 0.f, 0.f, 0.f};

        for (int kc = 0; kc < nK4; ++kc) {
            // ISA 16x16x4 f32 layout: VGPR r, lane-half h -> K = 4*kc + 2*h + r
            const int kb = (kc << 2) + (half << 1);
            v2f a, bb;
            a.x  = Ash[(ti * 16 + n16) * Kp + kb + 0];   // A[m=n16+16ti][k]
            a.y  = Ash[(ti * 16 + n16) * Kp + kb + 1];
            bb.x = Ysh[(kb + 0) * ncols + tj * 16 + n16]; // B[k][n=n16+16tj]
            bb.y = Ysh[(kb + 1) * ncols + tj * 16 + n16];
            c = __builtin_amdgcn_wmma_f32_16x16x4_f32(
                    /*neg_a=*/false, a, /*neg_b=*/false, bb,
                    /*c_mod=*/(short)0, c, /*reuse_a=*/false, /*reuse_b=*/false);
        }

        // D layout: VGPR v, half h -> row = 16*ti + v + 8*h, col = 16*tj + n16
        const int rbase = ti * 16 + (half << 3);
        const int col   = tj * 16 + n16;
#pragma unroll
        for (int v = 0; v < 8; ++v) {
            out[obase + (size_t)(rbase + v) * ncols + col] = c[v];
        }
    }
}

extern "C" void kernel_launch(void* const* d_in, const int* in_sizes, int n_in,
                              void* d_out, int out_size, void* d_ws, size_t ws_size,
                              hipStream_t stream) {
    (void)in_sizes; (void)out_size; (void)d_ws; (void)ws_size;

    KParams P;
    for (int l = 0; l < 6; ++l) P.x[l] = (const float*)d_in[l];

    // Enumerate paths exactly like the reference's _paths()
    int pathL[NPATHS], pathl1[NPATHS], pathl2[NPATHS];
    int nPperL[6] = {0, 0, 0, 0, 0, 0};
    int np = 0;
    for (int l1 = 0; l1 <= L_MAX; ++l1)
        for (int l2 = l1; l2 <= L_MAX; ++l2) {
            const int Lmax = (l1 + l2 < L_MAX) ? (l1 + l2) : L_MAX;
            for (int L = l2 - l1; L <= Lmax; ++L) {
                pathL[np] = L; pathl1[np] = l1; pathl2[np] = l2;
                nPperL[L]++; np++;
            }
        }
    // np == 69

    // Output base offset of each out[L] slab (flat concat of the returned tuple)
    unsigned long long outBaseL[6];
    unsigned long long acc = 0;
    for (int L = 0; L < 6; ++L) {
        outBaseL[L] = acc;
        acc += (unsigned long long)BATCH * 4096ull * (unsigned long long)nPperL[L] * (2ull * L + 1ull);
    }

    // CG offsets in case the harness concatenated the cgs dict into one input
    unsigned long long cgOff[NPATHS];
    unsigned long long cacc = 0;
    for (int pI = 0; pI < NPATHS; ++pI) {
        cgOff[pI] = cacc;
        cacc += (unsigned long long)(2 * pathL[pI] + 1) * (2 * pathl2[pI] + 1) * (2 * pathl1[pI] + 1);
    }

    int pCountL[6] = {0, 0, 0, 0, 0, 0};
    for (int pI = 0; pI < NPATHS; ++pI) {
        const int L = pathL[pI], l1 = pathl1[pI], l2 = pathl2[pI];
        P.meta[pI]    = L | (l1 << 4) | (l2 << 8);
        P.bstride[pI] = 4096u * (unsigned int)nPperL[L] * (unsigned int)(2 * L + 1);
        P.pbase[pI]   = outBaseL[L] + (unsigned long long)pCountL[L] * 4096ull * (2ull * L + 1ull);
        pCountL[L]++;
        if (n_in >= 6 + NPATHS) {
            P.cg[pI] = (const float*)d_in[6 + pI];            // per-tensor flattened dict
        } else {
            P.cg[pI] = (const float*)d_in[6] + cgOff[pI];     // single concatenated blob
        }
    }

    dim3 grid(NPATHS, BATCH, 1);
    cg_tensorprod_wmma<<<grid, 256, 0, stream>>>(P, (float*)d_out);
}